// Coupling_36026185679231
// MI455X (gfx1250) — compile-verified
//
#include <hip/hip_runtime.h>
#include <stdint.h>

// Problem constants (from reference): B=16, K=8, C=64, L=2048
#define B_ 16
#define K_ 8
#define C_ 64
#define L_ 2048
#define CL4 ((C_ * L_) / 4)           // 32768 float4 per batch per plane
#define BLOCK 256                     // 8 waves (wave32)
#define BPB 32                        // blocks per batch
#define ITERS (CL4 / (BPB * BLOCK))   // 4 float4-groups per thread

#define OFFSET_F 2.0f
#define SIG_OFF 0.11920292202211755f      // 1 - sigmoid(2.0)
#define LOG_EPS -16.11809565095832f       // log(1e-7)
#define LOG_1ME -1.0000000050000001e-7f   // log(1 - 1e-7)

// ---- CDNA5 async global->LDS path (builtin if present, inline asm fallback) ----
#if __has_builtin(__builtin_amdgcn_s_wait_asynccnt)
#define WAIT_ASYNC(n) __builtin_amdgcn_s_wait_asynccnt(n)
#else
#define WAIT_ASYNC(n) asm volatile("s_wait_asynccnt %0" ::"n"(n))
#endif

// Pointee type per hipcc diagnostic: int __attribute__((vector_size(16)))
typedef int v4i_t __attribute__((vector_size(16)));
typedef __attribute__((address_space(1))) v4i_t* gptr_b128;
typedef __attribute__((address_space(3))) v4i_t* lptr_b128;

__device__ __forceinline__ void async_copy16(const void* g, void* l) {
#if __has_builtin(__builtin_amdgcn_global_load_async_to_lds_b128)
  __builtin_amdgcn_global_load_async_to_lds_b128(
      (gptr_b128)(uintptr_t)g,
      (lptr_b128)(uint32_t)(uintptr_t)l,   // generic low 32 bits == LDS offset
      /*offset=*/0, /*cpol=*/0);
#else
  uint32_t lo = (uint32_t)(uintptr_t)l;
  uint64_t ga = (uint64_t)(uintptr_t)g;
  asm volatile("global_load_async_to_lds_b128 %0, %1, off" ::"v"(lo), "v"(ga)
               : "memory");
#endif
}

// Streaming logsumexp accumulator update (branchless, 2x v_exp)
#define LSE_UPD(m, sacc, v)                                  \
  {                                                          \
    float _v = (v);                                          \
    float _mn = fmaxf((m), _v);                              \
    (sacc) = (sacc) * __expf((m) - _mn) + __expf(_v - _mn);  \
    (m) = _mn;                                               \
  }

__device__ __forceinline__ float comp(const float4& v, int c) {
  switch (c) { case 0: return v.x; case 1: return v.y; case 2: return v.z; default: return v.w; }
}
__device__ __forceinline__ void setcomp(float4& v, int c, float f) {
  switch (c) { case 0: v.x = f; break; case 1: v.y = f; break; case 2: v.z = f; break; default: v.w = f; }
}

__global__ __launch_bounds__(BLOCK) void coupling_fused(
    const float* __restrict__ xg, const float* __restrict__ ag,
    const float* __restrict__ bg, const float* __restrict__ pig,
    const float* __restrict__ mug, const float* __restrict__ sg,
    float* __restrict__ outg, float* __restrict__ ws) {
  // Double-buffered LDS staging: [buf][pi/mu/s][tid] -> 24 KB
  __shared__ float4 stage[2][3][BLOCK];
  __shared__ float red[BLOCK];

  const int tid = threadIdx.x;
  const int bIdx = blockIdx.x / BPB;   // batch
  const int blk = blockIdx.x % BPB;    // block within batch

  const float4* x4 = (const float4*)xg + (size_t)bIdx * CL4;
  const float4* a4 = (const float4*)ag + (size_t)bIdx * CL4;
  const float4* b4 = (const float4*)bg + (size_t)bIdx * CL4;
  float4* o4 = (float4*)outg + (size_t)bIdx * CL4;
  const float4* pi4 = (const float4*)pig + (size_t)bIdx * ((size_t)K_ * CL4);
  const float4* mu4 = (const float4*)mug + (size_t)bIdx * ((size_t)K_ * CL4);
  const float4* s4  = (const float4*)sg  + (size_t)bIdx * ((size_t)K_ * CL4);

  float local = 0.0f;

  for (int it = 0; it < ITERS; ++it) {
    const size_t i4 = (size_t)blk * (ITERS * BLOCK) + (size_t)it * BLOCK + tid;

    const float4 xv = x4[i4];
    const float4 av = a4[i4];
    const float4 bv = b4[i4];

    // Three streaming logsumexps over K (per float4 component):
    //  P: lse(pi)   -> log-softmax normalizer
    //  1: lse(pi + logsigmoid(z))  -> mixture log-CDF (+logZ)
    //  2: lse(pi + z - s - 2*softplus(z)) -> mixture log-PDF (+logZ)
    float mP[4], sP[4], m1[4], s1[4], m2[4], s2[4];
#pragma unroll
    for (int c = 0; c < 4; ++c) {
      mP[c] = m1[c] = m2[c] = -1e30f;
      sP[c] = s1[c] = s2[c] = 0.0f;
    }

    // Prime pipeline: stage k=0 into buffer 0
    async_copy16(pi4 + i4, &stage[0][0][tid]);
    async_copy16(mu4 + i4, &stage[0][1][tid]);
    async_copy16(s4  + i4, &stage[0][2][tid]);

#pragma unroll
    for (int k = 0; k < K_; ++k) {
      const int buf = k & 1;
      if (k + 1 < K_) {
        const size_t j4 = i4 + (size_t)(k + 1) * CL4;
        async_copy16(pi4 + j4, &stage[buf ^ 1][0][tid]);
        async_copy16(mu4 + j4, &stage[buf ^ 1][1][tid]);
        async_copy16(s4  + j4, &stage[buf ^ 1][2][tid]);
        WAIT_ASYNC(3);   // next stage in flight; current stage complete
      } else {
        WAIT_ASYNC(0);   // last stage: drain
      }
      asm volatile("" ::: "memory");
      const float4 pv = stage[buf][0][tid];
      const float4 mv = stage[buf][1][tid];
      const float4 sv = stage[buf][2][tid];
#pragma unroll
      for (int c = 0; c < 4; ++c) {
        const float p  = comp(pv, c);
        const float mm = comp(mv, c);
        const float ss = comp(sv, c);
        const float xc = comp(xv, c);
        const float e = __expf(-ss);
        const float z = (xc - mm) * e;
        const float az = fabsf(z);
        const float Lg = __logf(1.0f + __expf(-az));  // log1p(exp(-|z|))
        const float lsig = fminf(z, 0.0f) - Lg;       // log sigmoid(z)
        const float sp   = fmaxf(z, 0.0f) + Lg;       // softplus(z)
        LSE_UPD(mP[c], sP[c], p);
        LSE_UPD(m1[c], s1[c], p + lsig);
        LSE_UPD(m2[c], s2[c], p + z - ss - 2.0f * sp);
      }
    }

    float4 ov;
#pragma unroll
    for (int c = 0; c < 4; ++c) {
      const float logZ = mP[c] + __logf(sP[c]);
      const float lcdf = m1[c] + __logf(s1[c]) - logZ;
      const float lpdf = m2[c] + __logf(s2[c]) - logZ;
      const float lu = fminf(fmaxf(lcdf, LOG_EPS), LOG_1ME);  // log(clamped u)
      const float u = __expf(lu);
      const float l1mu = log1pf(-u);
      const float y = lu - l1mu;
      const float scale_ldj = -lu - l1mu;
      const float sig = __builtin_amdgcn_rcpf(1.0f + __expf(-(comp(av, c) + OFFSET_F)));
      const float scale = sig + SIG_OFF;
      setcomp(ov, c, (y + comp(bv, c)) * scale);
      local += lpdf + scale_ldj + __logf(scale);
    }
    o4[i4] = ov;
  }

  // Deterministic per-block reduction of sldj contributions
  red[tid] = local;
  __syncthreads();
#pragma unroll
  for (int off = BLOCK / 2; off > 0; off >>= 1) {
    if (tid < off) red[tid] += red[tid + off];
    __syncthreads();
  }
  if (tid == 0) ws[blockIdx.x] = red[0];
}

// Stage 2: fold per-block partials (fixed order) + sldj input
__global__ __launch_bounds__(BPB) void sldj_reduce(
    const float* __restrict__ ws, const float* __restrict__ sldj_in,
    float* __restrict__ out_sldj) {
  __shared__ float sm[BPB];
  const int b = blockIdx.x, t = threadIdx.x;
  sm[t] = ws[b * BPB + t];
  __syncthreads();
#pragma unroll
  for (int off = BPB / 2; off > 0; off >>= 1) {
    if (t < off) sm[t] += sm[t + off];
    __syncthreads();
  }
  if (t == 0) out_sldj[b] = sldj_in[b] + sm[0];
}

extern "C" void kernel_launch(void* const* d_in, const int* in_sizes, int n_in,
                              void* d_out, int out_size, void* d_ws, size_t ws_size,
                              hipStream_t stream) {
  (void)in_sizes; (void)n_in; (void)out_size; (void)ws_size;
  const float* x    = (const float*)d_in[0];  // x_change [B,C,L]
  const float* a    = (const float*)d_in[1];  // a        [B,C,L]
  const float* b    = (const float*)d_in[2];  // b        [B,C,L]
  const float* pi   = (const float*)d_in[3];  // pi       [B,K,C,L]
  const float* mu   = (const float*)d_in[4];  // mu       [B,K,C,L]
  const float* s    = (const float*)d_in[5];  // s        [B,K,C,L]
  const float* sldj = (const float*)d_in[6];  // sldj     [B]

  float* out = (float*)d_out;                                   // [B,C,L]
  float* out_sldj = out + (size_t)B_ * C_ * L_;                 // [B]
  float* ws = (float*)d_ws;                                     // [B*BPB] partials

  coupling_fused<<<dim3(B_ * BPB), dim3(BLOCK), 0, stream>>>(x, a, b, pi, mu, s,
                                                             out, ws);
  sldj_reduce<<<dim3(B_), dim3(BPB), 0, stream>>>(ws, sldj, out_sldj);
}